// TreeRCNN_63187558859086
// MI455X (gfx1250) — compile-verified
//
#include <hip/hip_runtime.h>
#include <stdint.h>

// Point-in-box fixed-capacity query + ordered gather (TreeRCNN-style).
//   points : [N,3] f32, anchors : [A,6] f32 (cx,cy,cz,w,l,h), n_max scalar i32
//   out    : [A, n_max, 3] f32 (centered by (cx,cy,0), zero padded)
//   counts : [A] i32   (appended flat after out)
//
// MI455X strategy:
//  * points (2.4MB) tiled through LDS with CDNA5 async global->LDS B128
//    copies (ASYNCcnt path), double-buffered so DMA overlaps compute.
//  * 16 anchor-waves per 512-thread block share each staged tile
//    (16x reuse of L2-resident point data).
//  * wave32 order-preserving compaction: ballot_w32 + mbcnt_lo gives each
//    hit its slot; matches jax.lax.top_k ascending-index tie order.
//  * full tiles take a clamp-free unrolled fast path; only the last partial
//    tile pays bounds predicates.
//  * zero padding + counts written by the same kernel: each output element
//    is stored exactly once.

#define TILE_PTS 2048
#define TILE_DW  (TILE_PTS * 3)     // 6144 dwords = 24KB per buffer
#define TILE_Q   (TILE_DW / 4)      // 1536 x b128 transfers per tile
#define BLOCK    512
#define WAVES    (BLOCK / 32)       // 16 anchor-waves per block

__global__ void __launch_bounds__(BLOCK)
pib_scan_kernel(const float* __restrict__ points,
                const float* __restrict__ anchors,
                const int*   __restrict__ nmax_ptr,
                float*       __restrict__ out,
                int N, int A)
{
    __shared__ __align__(16) float tile[2][TILE_DW];

    const int      n_max = *nmax_ptr;
    const unsigned lane  = __builtin_amdgcn_mbcnt_lo(~0u, 0u);  // wave32 lane id
    const int      wave  = threadIdx.x >> 5;
    const int      a     = blockIdx.x * WAVES + wave;

    // Per-wave (uniform) anchor params. Same float ops as reference:
    // bounds = c -/+ w*0.5f (exact), inclusive compares, z in [0,h].
    float cx = 0.f, cy = 0.f, xmin = 0.f, xmax = 0.f, ymin = 0.f, ymax = 0.f, h = 0.f;
    if (a < A) {
        cx = anchors[a * 6 + 0];
        cy = anchors[a * 6 + 1];
        const float w = anchors[a * 6 + 3];
        const float l = anchors[a * 6 + 4];
        h  = anchors[a * 6 + 5];
        xmin = cx - w * 0.5f;  xmax = cx + w * 0.5f;
        ymin = cy - l * 0.5f;  ymax = cy + l * 0.5f;
    }

    // Low 32 bits of a generic LDS pointer == LDS byte offset.
    const uint32_t lds_base = (uint32_t)(uintptr_t)(&tile[0][0]);
    const uint64_t gbase    = (uint64_t)(uintptr_t)points;
    const int      total_q  = (N * 3) >> 2;   // N*3 divisible by 4 here; guarded below

    // Issue async B128 global->LDS copies for the tile starting at tstart.
    auto issue_tile = [&](int tstart, int buf) {
        if (tstart >= N) return;
        const int      base_q = (tstart * 3) >> 2;          // tile starts 16B-aligned
        const uint32_t lbase  = lds_base + (uint32_t)buf * (uint32_t)(TILE_DW * 4);
        for (int q = threadIdx.x; q < TILE_Q; q += BLOCK) {
            if (base_q + q < total_q) {
                const uint32_t lds_addr = lbase + (uint32_t)q * 16u;
                const uint64_t gaddr    = gbase + (uint64_t)(base_q + q) * 16u;
                asm volatile("global_load_async_to_lds_b128 %0, %1, off"
                             :: "v"(lds_addr), "v"(gaddr) : "memory");
            }
        }
    };

    // One 32-point compaction step; returns nothing, updates cnt by reference.
    float* const obase = out + (size_t)(a < A ? a : 0) * (size_t)n_max * 3;
    auto step = [&](const float* __restrict__ tl, int j, bool pred, int& cnt) {
        const float px = tl[j * 3 + 0];
        const float py = tl[j * 3 + 1];
        const float pz = tl[j * 3 + 2];
        const bool in = pred &
                        (px >= xmin) & (px <= xmax) &
                        (py >= ymin) & (py <= ymax) &
                        (pz >= 0.0f) & (pz <= h);
        // Order-preserving compaction: slot = cnt + #hits in lower lanes.
        const unsigned bal    = __builtin_amdgcn_ballot_w32(in);
        const int      prefix = (int)__builtin_amdgcn_mbcnt_lo(bal, 0u);
        const int      slot   = cnt + prefix;
        if (in && (slot < n_max)) {
            float* o = obase + (size_t)slot * 3;
            o[0] = px - cx;   // offset = (cx, cy, 0)
            o[1] = py - cy;
            o[2] = pz;
        }
        cnt += __popc(bal);
    };

    // Prologue: stage tile 0 into buffer 0.
    issue_tile(0, 0);
    asm volatile("s_wait_asynccnt 0" ::: "memory");
    __syncthreads();

    int cnt = 0;   // running hit count for this wave's anchor
    int buf = 0;

    for (int tstart = 0; tstart < N; tstart += TILE_PTS) {
        // Prefetch next tile into the other buffer (overlaps compute below).
        issue_tile(tstart + TILE_PTS, buf ^ 1);

        const int tile_n = (N - tstart < TILE_PTS) ? (N - tstart) : TILE_PTS;
        const float* __restrict__ tl = &tile[buf][0];

        if (a < A) {
            if (tile_n == TILE_PTS) {
                // Fast path: full tile, no clamp / no bounds predicate.
                #pragma unroll 4
                for (int j0 = 0; j0 < TILE_PTS; j0 += 32)
                    step(tl, j0 + (int)lane, true, cnt);
            } else {
                // Remainder tile: predicate on j < tile_n, clamp LDS index.
                for (int j0 = 0; j0 < tile_n; j0 += 32) {
                    const int j = j0 + (int)lane;
                    step(tl, (j < tile_n) ? j : 0, j < tile_n, cnt);
                }
            }
        }

        // Wait for this wave's prefetch, then block-wide barrier: everyone is
        // done reading tile[buf], and tile[buf^1] is fully resident.
        asm volatile("s_wait_asynccnt 0" ::: "memory");
        __syncthreads();
        buf ^= 1;
    }

    if (a < A) {
        // Zero-fill the unmatched tail slots (reference zero-pads).
        const int start = ((cnt < n_max) ? cnt : n_max) * 3;
        const int end   = n_max * 3;
        for (int i = start + (int)lane; i < end; i += 32)
            obase[i] = 0.0f;
        if (lane == 0) {
            int* counts = (int*)(out + (size_t)A * (size_t)n_max * 3);
            counts[a] = cnt;
        }
    }
}

extern "C" void kernel_launch(void* const* d_in, const int* in_sizes, int n_in,
                              void* d_out, int out_size, void* d_ws, size_t ws_size,
                              hipStream_t stream) {
    const float* points  = (const float*)d_in[0];
    const float* anchors = (const float*)d_in[1];
    const int*   nmax    = (const int*)d_in[2];
    float*       out     = (float*)d_out;

    const int N = in_sizes[0] / 3;   // 200000
    const int A = in_sizes[1] / 6;   // 2048

    // One wave per anchor; 16 anchors per block share double-buffered LDS tiles.
    const int grid = (A + WAVES - 1) / WAVES;
    pib_scan_kernel<<<grid, BLOCK, 0, stream>>>(points, anchors, nmax, out, N, A);
}